// SafenessLoss_22462678958338
// MI455X (gfx1250) — compile-verified
//
#include <hip/hip_runtime.h>

// CDNA5 / gfx1250, wave32.
// Safeness triplet-style loss: bandwidth-bound (541 MB @ 23.3 TB/s ~ 23 us floor).
// Distances computed as diagonals of per-tile Gram matrices via
// V_WMMA_F32_16X16X4_F32 (fp32 exact, symmetric operand shared A==B).

typedef __attribute__((ext_vector_type(2))) float v2f;
typedef __attribute__((ext_vector_type(8))) float v8f;

#define KSTREAMS 8
#define BATCH    131072
#define DIM      128
#define NTILES   (BATCH / 16)   // 8192 tiles of 16 samples

__global__ __launch_bounds__(256) void safeness_main(
    const float* __restrict__ emb,     // [K, B, D] fp32
    const int*   __restrict__ target,  // [K, B] int32
    float*       __restrict__ accum)   // accum[0]=loss sum, accum[1]=count
{
    const int lane = threadIdx.x & 31;
    const int wave = threadIdx.x >> 5;
    const int wavesPerBlock = blockDim.x >> 5;
    const int waveId = blockIdx.x * wavesPerBlock + wave;
    const int nWaves = gridDim.x * wavesPerBlock;

    // WMMA A(16x4 f32) operand layout (wave32):
    //   lanes 0-15 : row m=lane,    K = d0+0 (v0), d0+1 (v1)
    //   lanes 16-31: row m=lane-16, K = d0+2 (v0), d0+3 (v1)
    // B(4x16) layout for B = A^T is elementwise identical -> share registers.
    const int m  = lane & 15;
    const int cg = (lane >> 4) << 1;           // 0 or 2

    const size_t strideK = (size_t)BATCH * DIM; // 2^24 elements per stream

    float lossAcc = 0.f, cntAcc = 0.f;

    for (int tile = waveId; tile < NTILES; tile += nWaves) {
        const int b0 = tile << 4;
        const size_t rowBase = (size_t)(b0 + m) * DIM + cg;

        v8f acc[7];
        #pragma unroll
        for (int k = 0; k < 7; ++k) { v8f z = {}; acc[k] = z; }

        const float* a_ptr = emb + rowBase;

        #pragma unroll 4
        for (int d0 = 0; d0 < DIM; d0 += 4) {
            const v2f a2 = *(const v2f*)(a_ptr + d0);
            #pragma unroll
            for (int k = 0; k < 7; ++k) {
                const float* e_ptr = emb + (size_t)(k + 1) * strideK + rowBase;
                const v2f e2 = *(const v2f*)(e_ptr + d0);
                const v2f df = a2 - e2;
                // G_k += Diff(16x4) x Diff^T(4x16); diagonal = squared distances
                acc[k] = __builtin_amdgcn_wmma_f32_16x16x4_f32(
                    false, df, false, df, (short)0, acc[k], false, false);
            }
        }

        // ---- diagonal extraction ----
        // G[i,i]: i<8  -> lane i,    VGPR i
        //         i>=8 -> lane i+16, VGPR i-8
        const int vsel = (lane < 16) ? lane : (lane - 24);
        float dk[7];
        #pragma unroll
        for (int k = 0; k < 7; ++k) {
            float d = acc[k][0];
            #pragma unroll
            for (int r = 1; r < 8; ++r) d = (vsel == r) ? acc[k][r] : d;
            dk[k] = d;
        }
        const bool active = (lane < 8) || (lane >= 24);     // diagonal holders
        const int  i      = (lane < 16) ? lane : (lane - 16);
        const int  b      = b0 + i;

        // ---- per-sample triplet logic (branchless) ----
        const int t0 = target[b];
        float same_sum = 0.f;
        float mx = -__builtin_inff();
        bool any_same = false;
        bool same[7];
        #pragma unroll
        for (int k = 0; k < 7; ++k) {
            const bool s = (target[(size_t)(k + 1) * BATCH + b] == t0);
            same[k] = s;
            any_same = any_same || s;
            same_sum += s ? dk[k] : 0.f;
            mx = s ? fmaxf(mx, dk[k]) : mx;
        }
        const float alpha = any_same ? mx : 1.0f;
        float neg_sum = 0.f;
        bool any_neg = false;
        #pragma unroll
        for (int k = 0; k < 7; ++k) {
            const float mg = dk[k] - alpha;
            const bool nm = (!same[k]) && (mg < 0.f);
            neg_sum += nm ? mg : 0.f;
            any_neg = any_neg || nm;
        }
        const bool include = any_same || any_neg;
        const float loss = same_sum - neg_sum;
        lossAcc += (active && include) ? loss : 0.f;
        cntAcc  += (active && include) ? 1.f  : 0.f;
    }

    // ---- wave32 butterfly reduction ----
    #pragma unroll
    for (int s = 16; s >= 1; s >>= 1) {
        lossAcc += __shfl_xor(lossAcc, s, 32);
        cntAcc  += __shfl_xor(cntAcc,  s, 32);
    }

    __shared__ float sL[8];
    __shared__ float sC[8];
    if (lane == 0) { sL[wave] = lossAcc; sC[wave] = cntAcc; }
    __syncthreads();
    if (threadIdx.x == 0) {
        float L = 0.f, C = 0.f;
        for (int w = 0; w < wavesPerBlock; ++w) { L += sL[w]; C += sC[w]; }
        atomicAdd(&accum[0], L);
        atomicAdd(&accum[1], C);
    }
}

__global__ void safeness_init(float* accum) {
    accum[0] = 0.f;
    accum[1] = 0.f;
}

__global__ void safeness_final(const float* __restrict__ accum,
                               float* __restrict__ out) {
    out[0] = accum[0] / fmaxf(accum[1], 1.0f);
}

extern "C" void kernel_launch(void* const* d_in, const int* in_sizes, int n_in,
                              void* d_out, int out_size, void* d_ws, size_t ws_size,
                              hipStream_t stream) {
    const float* emb    = (const float*)d_in[0];
    const int*   target = (const int*)d_in[1];
    float*       accum  = (float*)d_ws;
    float*       out    = (float*)d_out;

    hipLaunchKernelGGL(safeness_init, dim3(1), dim3(1), 0, stream, accum);
    // 1024 blocks x 8 waves = 8192 waves -> one 16-sample tile per wave
    hipLaunchKernelGGL(safeness_main, dim3(1024), dim3(256), 0, stream,
                       emb, target, accum);
    hipLaunchKernelGGL(safeness_final, dim3(1), dim3(1), 0, stream, accum, out);
}